// DeepChebNet_76433238000372
// MI455X (gfx1250) — compile-verified
//
#include <hip/hip_runtime.h>
#include <math.h>

// ---------------------------------------------------------------------------
// DeepChebNet for MI455X (gfx1250): fp32 WMMA GEMMs + L2-resident edge scatter
// GEMM v2: full weight tensor resident in 320KB WGP LDS; each wave keeps all
// 8 column tiles in registers -> 24 WMMAs per 3 global A-loads per k-step.
// ---------------------------------------------------------------------------

typedef __attribute__((ext_vector_type(2))) float v2f;
typedef __attribute__((ext_vector_type(8))) float v8f;

#define NN 50000
#define NE 800000
#define DD 128
#define BN_EPS 1e-5f

// ---------------- elementwise / graph-normalization kernels ----------------

__global__ __launch_bounds__(256) void zero_f4(float4* __restrict__ p, int n4) {
  int i = blockIdx.x * blockDim.x + threadIdx.x;
  if (i < n4) p[i] = make_float4(0.f, 0.f, 0.f, 0.f);
}

__global__ __launch_bounds__(256) void deg_scatter(const int* __restrict__ src,
                                                   const float* __restrict__ w,
                                                   float* __restrict__ deg, int ne) {
  int e = blockIdx.x * blockDim.x + threadIdx.x;
  if (e < ne) atomicAdd(deg + src[e], w[e]);
}

__global__ __launch_bounds__(256) void dinv_k(const float* __restrict__ deg,
                                              float* __restrict__ dinv, int n) {
  int i = blockIdx.x * blockDim.x + threadIdx.x;
  if (i < n) {
    float d = deg[i];
    dinv[i] = (d > 0.f) ? rsqrtf(d) : 0.f;
  }
}

__global__ __launch_bounds__(256) void norm_k(const int* __restrict__ src,
                                              const int* __restrict__ dst,
                                              const float* __restrict__ w,
                                              const float* __restrict__ dinv,
                                              float* __restrict__ nrm, int ne) {
  int e = blockIdx.x * blockDim.x + threadIdx.x;
  if (e < ne) nrm[e] = -dinv[src[e]] * w[e] * dinv[dst[e]];
}

// One wave per edge: lane l handles features [4l, 4l+4). Gather of x[src] is a
// fully-coalesced 512B row read; scatter is 4x global_atomic_add_f32 that
// resolve in the 192MB L2 (destination arrays are L2-resident).
__global__ __launch_bounds__(256) void prop_scatter(const float* __restrict__ x,
                                                    const int* __restrict__ src,
                                                    const int* __restrict__ dst,
                                                    const float* __restrict__ nrm,
                                                    float* __restrict__ out, int ne) {
  int t = blockIdx.x * blockDim.x + threadIdx.x;
  int e = t >> 5;
  if (e >= ne) return;
  int f = (t & 31) * 4;
  float w = nrm[e];
  const float4 xv = *(const float4*)(x + (long)src[e] * DD + f);
  float* o = out + (long)dst[e] * DD + f;
  atomicAdd(o + 0, w * xv.x);
  atomicAdd(o + 1, w * xv.y);
  atomicAdd(o + 2, w * xv.z);
  atomicAdd(o + 3, w * xv.w);
}

// ---------------------------- WMMA GEMM kernel -----------------------------
// out = A0@W[0] (+ A1@W[1] + A2'@W[2]) + bias, with A2' = 2*A2 - A0 (Cheb).
// EPI: 0 = bias only, 1 = bias+ReLU, 2 = bias+BN+ReLU.
//
// Block = 256 threads = 8 waves; block covers 128 rows x all 128 cols.
// ALL NMAT weight matrices are staged into LDS once (NMAT=3: 3*128*136*4 =
// 208,896 B, fits the 320KB WGP LDS). Each wave owns a 16-row strip and keeps
// all 8 16x16 accumulators live (64 VGPRs), so one pass over K issues
// NMAT*8 WMMAs per NMAT global A-loads -> 8x less A traffic, deep independent
// WMMA chains (no RAW hazards between acc[jt]).
//
// LDS stride 136 floats: lanes 16-31 read K+2 (bank shift 2*136 mod 64 = 16),
// so the two half-waves hit disjoint 16-bank groups -> conflict-free.
//
// f32 WMMA operand layout (ISA 7.12.2): A 16x4: lanes0-15 K={0,1}, lanes16-31
// K={2,3}; B 4x16 mirrored over N; C/D: vgpr r -> M = r + 8*(lane>=16).
template <int NMAT, int EPI, bool CHEB>
__global__ __launch_bounds__(256) void gemm_wmma(
    const float* __restrict__ A0, const float* __restrict__ A1,
    const float* __restrict__ A2, const float* __restrict__ W,
    const float* __restrict__ bias, const float* __restrict__ bn_g,
    const float* __restrict__ bn_b, const float* __restrict__ bn_mu,
    const float* __restrict__ bn_var, float* __restrict__ out, int n) {
  constexpr int BST = DD + 8;  // 136-float row stride (see bank note above)
  __shared__ float Bs[NMAT][DD][BST];

  const int tid = threadIdx.x;
  // stage the whole weight tensor into LDS (coalesced 512B global rows)
  for (int idx = tid; idx < NMAT * DD * DD; idx += 256) {
    int mat = idx >> 14;       // / (128*128)
    int rem = idx & 16383;
    Bs[mat][rem >> 7][rem & 127] = W[idx];
  }
  __syncthreads();

  const int lane = tid & 31;
  const int wave = tid >> 5;
  const int m0   = blockIdx.x * 128 + wave * 16;
  const int mrow = m0 + (lane & 15);
  const long arow = (long)((mrow < n) ? mrow : 0) * DD;  // clamp: EXEC stays all-1s
  const int koff = (lane >> 4) << 1;  // 0 for lanes 0-15, 2 for lanes 16-31
  const int c    = lane & 15;

  v8f acc[8];
  const v8f vz = {0.f, 0.f, 0.f, 0.f, 0.f, 0.f, 0.f, 0.f};
#pragma unroll
  for (int jt = 0; jt < 8; ++jt) acc[jt] = vz;

  for (int kb = 0; kb < DD; kb += 4) {
    const int k = kb + koff;
    v2f a0 = *(const v2f*)(A0 + arow + k);
    v2f a1, a2;
    if (NMAT >= 2) a1 = *(const v2f*)(A1 + arow + k);
    if (NMAT >= 3) {
      v2f p = *(const v2f*)(A2 + arow + k);
      if (CHEB) {  // Chebyshev recurrence fused into the A-operand
        a2.x = 2.f * p.x - a0.x;
        a2.y = 2.f * p.y - a0.y;
      } else {
        a2 = p;
      }
    }
#pragma unroll
    for (int jt = 0; jt < 8; ++jt) {
      const int cc = jt * 16 + c;
      v2f b0;
      b0.x = Bs[0][k][cc];
      b0.y = Bs[0][k + 1][cc];
      acc[jt] = __builtin_amdgcn_wmma_f32_16x16x4_f32(false, a0, false, b0,
                                                      (short)0, acc[jt], false, false);
      if (NMAT >= 2) {
        v2f b1;
        b1.x = Bs[1][k][cc];
        b1.y = Bs[1][k + 1][cc];
        acc[jt] = __builtin_amdgcn_wmma_f32_16x16x4_f32(false, a1, false, b1,
                                                        (short)0, acc[jt], false, false);
      }
      if (NMAT >= 3) {
        v2f b2;
        b2.x = Bs[2][k][cc];
        b2.y = Bs[2][k + 1][cc];
        acc[jt] = __builtin_amdgcn_wmma_f32_16x16x4_f32(false, a2, false, b2,
                                                        (short)0, acc[jt], false, false);
      }
    }
  }

  // fused epilogue + store (64B-contiguous per half-wave)
  const int rbase = m0 + ((lane >> 4) << 3);
#pragma unroll
  for (int jt = 0; jt < 8; ++jt) {
    const int cn = jt * 16 + c;
    float bsc = bias[cn];
    float scale = 1.f, shift = 0.f;
    if (EPI == 2) {
      scale = rsqrtf(bn_var[cn] + BN_EPS) * bn_g[cn];
      shift = bn_b[cn] - bn_mu[cn] * scale;
    }
#pragma unroll
    for (int r = 0; r < 8; ++r) {
      int row = rbase + r;
      float v = acc[jt][r] + bsc;
      if (EPI == 2) v = v * scale + shift;
      if (EPI >= 1) v = fmaxf(v, 0.f);
      if (row < n) out[(long)row * DD + cn] = v;
    }
  }
}

// ------------------------- final 128->1 head + sigmoid ---------------------
__global__ __launch_bounds__(256) void head_k(const float* __restrict__ h,
                                              const float* __restrict__ w2,
                                              const float* __restrict__ b2,
                                              float* __restrict__ out, int n) {
  int t = blockIdx.x * blockDim.x + threadIdx.x;
  int node = t >> 5;
  int lane = t & 31;
  if (node >= n) return;
  const float* row = h + (long)node * DD;
  float s = 0.f;
#pragma unroll
  for (int k = 0; k < DD; k += 32) s = fmaf(row[k + lane], w2[k + lane], s);
  for (int off = 16; off > 0; off >>= 1) s += __shfl_down(s, off, 32);
  if (lane == 0) out[node] = 1.f / (1.f + expf(-(s + b2[0])));
}

// --------------------------------- driver ----------------------------------

extern "C" void kernel_launch(void* const* d_in, const int* in_sizes, int n_in,
                              void* d_out, int out_size, void* d_ws, size_t ws_size,
                              hipStream_t stream) {
  const float* x    = (const float*)d_in[0];
  const int* ei     = (const int*)d_in[1];  // int32 (JAX default x64-off)
  const int* src    = ei;
  const int* dst    = ei + NE;
  const float* ew   = (const float*)d_in[2];
  const float* W_in  = (const float*)d_in[3];
  const float* b_in  = (const float*)d_in[4];
  const float* W_hid = (const float*)d_in[5];
  const float* b_hid = (const float*)d_in[6];
  const float* W_out = (const float*)d_in[7];
  const float* b_out = (const float*)d_in[8];
  const float* w1   = (const float*)d_in[9];
  const float* b1   = (const float*)d_in[10];
  const float* gam  = (const float*)d_in[11];
  const float* bet  = (const float*)d_in[12];
  const float* mu   = (const float*)d_in[13];
  const float* var  = (const float*)d_in[14];
  const float* w2   = (const float*)d_in[15];
  const float* b2   = (const float*)d_in[16];
  float* outp = (float*)d_out;

  char* wsb = (char*)d_ws;
  size_t off = 0;
  auto alloc = [&](size_t bytes) -> float* {
    float* p = (float*)(wsb + off);
    off += (bytes + 255) & ~(size_t)255;
    return p;
  };
  float* deg  = alloc((size_t)NN * 4);
  float* dinv = alloc((size_t)NN * 4);
  float* nrm  = alloc((size_t)NE * 4);
  float* tx1  = alloc((size_t)NN * DD * 4);
  float* pp   = alloc((size_t)NN * DD * 4);
  float* hA   = alloc((size_t)NN * DD * 4);
  float* hB   = alloc((size_t)NN * DD * 4);

  const int TB = 256;
  const int nodeBlocks = (NN + TB - 1) / TB;
  const int edgeBlocks = (NE + TB - 1) / TB;
  const int featF4     = NN * DD / 4;
  const int featZeroBl = (featF4 + TB - 1) / TB;
  const int degZeroBl  = ((NN + 3) / 4 + TB - 1) / TB;
  const int propBlocks = (NE * 32) / TB;        // 800000*32 / 256
  const int gemmBlocks = (NN + 127) / 128;      // 391
  const int headBlocks = (NN * 32 + TB - 1) / TB;

  // --- graph normalization (cheb_norm) ---
  zero_f4<<<degZeroBl, TB, 0, stream>>>((float4*)deg, (NN + 3) / 4);
  deg_scatter<<<edgeBlocks, TB, 0, stream>>>(src, ew, deg, NE);
  dinv_k<<<nodeBlocks, TB, 0, stream>>>(deg, dinv, NN);
  norm_k<<<edgeBlocks, TB, 0, stream>>>(src, dst, ew, dinv, nrm, NE);

  // --- one ChebConv layer: Tx1 = P(in), P = P(Tx1), GEMM(+ReLU) ---
  auto layer = [&](const float* in, const float* W, const float* b, float* o,
                   bool relu) {
    zero_f4<<<featZeroBl, TB, 0, stream>>>((float4*)tx1, featF4);
    prop_scatter<<<propBlocks, TB, 0, stream>>>(in, src, dst, nrm, tx1, NE);
    zero_f4<<<featZeroBl, TB, 0, stream>>>((float4*)pp, featF4);
    prop_scatter<<<propBlocks, TB, 0, stream>>>(tx1, src, dst, nrm, pp, NE);
    if (relu)
      gemm_wmma<3, 1, true><<<gemmBlocks, TB, 0, stream>>>(
          in, tx1, pp, W, b, nullptr, nullptr, nullptr, nullptr, o, NN);
    else
      gemm_wmma<3, 0, true><<<gemmBlocks, TB, 0, stream>>>(
          in, tx1, pp, W, b, nullptr, nullptr, nullptr, nullptr, o, NN);
  };

  layer(x,  W_in,  b_in,  hA, true);
  layer(hA, W_hid, b_hid, hB, true);
  layer(hB, W_out, b_out, hA, false);

  // --- MLP head: Linear + BN + ReLU (fused), then 128->1 + sigmoid ---
  gemm_wmma<1, 2, false><<<gemmBlocks, TB, 0, stream>>>(
      hA, nullptr, nullptr, w1, b1, gam, bet, mu, var, hB, NN);
  head_k<<<headBlocks, TB, 0, stream>>>(hB, w2, b2, outp, NN);
}